// GraphSAGE_71236327571567
// MI455X (gfx1250) — compile-verified
//
#include <hip/hip_runtime.h>

// ---------------------------------------------------------------------------
// GraphSAGE forward on MI455X (gfx1250, wave32, WMMA).
// Aggregation is memory/atomic bound (~3.3 GB through L2/HBM); GEMMs are tiny
// (~20 GFLOP) so we use fp32 WMMA (V_WMMA_F32_16X16X4_F32) for exactness.
// ---------------------------------------------------------------------------

typedef __attribute__((ext_vector_type(2))) float v2f;
typedef __attribute__((ext_vector_type(8))) float v8f;

#define N_NODES 50000
#define N_EDGES 800000
#define IN_DIM  256      // K for both layers (layer2 input is 256-dim h)
#define HID_DIM 256
#define OUT_DIM 128

// ---------------------------------------------------------------------------
__global__ void zero_f32(float* __restrict__ p, int n) {
    int i = blockIdx.x * blockDim.x + threadIdx.x;
    if (i < n) p[i] = 0.0f;
}

__global__ void degree_kernel(const int* __restrict__ dst,
                              float* __restrict__ deg, int n_edges) {
    int e = blockIdx.x * blockDim.x + threadIdx.x;
    if (e < n_edges) atomicAdd(&deg[dst[e]], 1.0f);
}

// deg[i] <- 1 / max(deg[i], 1)   (in place)
__global__ void invdeg_kernel(float* __restrict__ deg, int n) {
    int i = blockIdx.x * blockDim.x + threadIdx.x;
    if (i < n) deg[i] = 1.0f / fmaxf(deg[i], 1.0f);
}

// One thread per (edge, 4-float chunk): 64 chunks per edge (256 dims).
// agg[dst] += x[src]  (unnormalized sum; degree normalization fused in GEMM).
__global__ void scatter_kernel(const float* __restrict__ x,
                               const int* __restrict__ src,
                               const int* __restrict__ dst,
                               float* __restrict__ agg, int n_edges) {
    int gid = blockIdx.x * blockDim.x + threadIdx.x;
    int e = gid >> 6;          // edge index
    int c = gid & 63;          // float4 chunk within the 256-dim row
    if (e >= n_edges) return;
    const float4 v = *(const float4*)(x + (size_t)src[e] * IN_DIM + c * 4);
    float* a = agg + (size_t)dst[e] * IN_DIM + c * 4;
    atomicAdd(a + 0, v.x);
    atomicAdd(a + 1, v.y);
    atomicAdd(a + 2, v.z);
    atomicAdd(a + 3, v.w);
}

// ---------------------------------------------------------------------------
// Fused SAGE GEMM:  out = act( A_self @ W_self + (A_agg * invdeg) @ W_neigh + b )
//   A_self, A_agg : [N_NODES][256] row-major
//   W_self/W_neigh: [256][N] row-major, bias: [N]
// Block = 256 threads = 8 waves. Block owns N-panel tn (16 cols), stages both
// weight panels into LDS (2 x 256 x 16 f32 = 32 KB). Wave w computes the
// 16x16 tile at (blockIdx.y*8 + w, tn) using v_wmma_f32_16x16x4_f32.
//
// Fragment layouts per CDNA5 ISA 7.12.2 (wave32):
//   A 16x4 f32 (2 VGPR): lanes 0-15 -> M=lane, {K, K+1}; lanes 16-31 -> {K+2, K+3}
//   B 4x16 f32 (2 VGPR): lanes 0-15 -> N=lane, rows {K, K+1}; lanes 16-31 rows {K+2, K+3}
//   C/D 16x16 f32 (8 VGPR): vgpr g -> M = g + 8*(lane>=16), N = lane&15
// ---------------------------------------------------------------------------
template <int N, int DO_RELU>
__global__ __launch_bounds__(256)
void sage_gemm(const float* __restrict__ A_self,
               const float* __restrict__ A_agg,
               const float* __restrict__ invdeg,
               const float* __restrict__ W_self,
               const float* __restrict__ W_neigh,
               const float* __restrict__ bias,
               float* __restrict__ out) {
    __shared__ float Bs[2][IN_DIM][16];   // [self|neigh][K][panel col]  32 KB

    const int tn      = blockIdx.x;       // N-panel index
    const int wave    = threadIdx.x >> 5;
    const int lane    = threadIdx.x & 31;
    const int colbase = tn * 16;

    // Cooperative stage of both 256x16 weight panels (float4 per thread x4).
    for (int i = 0; i < 4; ++i) {
        int idx = threadIdx.x + i * 256;  // 0..1023 float4 slots per panel
        int r   = idx >> 2;
        int c4  = (idx & 3) * 4;
        *(float4*)&Bs[0][r][c4] = *(const float4*)&W_self [(size_t)r * N + colbase + c4];
        *(float4*)&Bs[1][r][c4] = *(const float4*)&W_neigh[(size_t)r * N + colbase + c4];
    }
    __syncthreads();

    const int tm = blockIdx.y * 8 + wave;           // M-tile index (wave-uniform)
    if (tm * 16 >= N_NODES) return;                 // 3125 tiles total

    const int halfsel = lane >> 4;                  // 0: K,K+1   1: K+2,K+3
    const int col     = lane & 15;
    const int mrow    = tm * 16 + col;              // A-matrix row for this lane
    const float idg   = invdeg[mrow];
    const float* arow_s = A_self + (size_t)mrow * IN_DIM;
    const float* arow_n = A_agg  + (size_t)mrow * IN_DIM;

    v8f c = {0.f, 0.f, 0.f, 0.f, 0.f, 0.f, 0.f, 0.f};

    for (int k = 0; k < IN_DIM; k += 4) {
        const int ka = k + (halfsel << 1);          // k or k+2 (8B aligned)
        v2f a_s = *(const v2f*)(arow_s + ka);       // global_load_b64
        v2f a_n = *(const v2f*)(arow_n + ka);
        a_n = a_n * idg;                            // fused mean normalization

        v2f b_s, b_n;
        b_s.x = Bs[0][ka][col];  b_s.y = Bs[0][ka + 1][col];
        b_n.x = Bs[1][ka][col];  b_n.y = Bs[1][ka + 1][col];

        c = __builtin_amdgcn_wmma_f32_16x16x4_f32(false, a_s, false, b_s,
                                                  (short)0, c, false, false);
        c = __builtin_amdgcn_wmma_f32_16x16x4_f32(false, a_n, false, b_n,
                                                  (short)0, c, false, false);
    }

    const float bv    = bias[colbase + col];
    const int   rbase = tm * 16 + (halfsel << 3);   // M = rbase + g
    for (int g = 0; g < 8; ++g) {
        float v = c[g] + bv;
        if (DO_RELU) v = fmaxf(v, 0.0f);
        out[(size_t)(rbase + g) * N + colbase + col] = v;
    }
}

// ---------------------------------------------------------------------------
extern "C" void kernel_launch(void* const* d_in, const int* in_sizes, int n_in,
                              void* d_out, int out_size, void* d_ws, size_t ws_size,
                              hipStream_t stream) {
    const float* in_feat = (const float*)d_in[0];
    const int*   esrc    = (const int*)  d_in[1];
    const int*   edst    = (const int*)  d_in[2];
    const float* W1s     = (const float*)d_in[3];
    const float* W1n     = (const float*)d_in[4];
    const float* b1      = (const float*)d_in[5];
    const float* W2s     = (const float*)d_in[6];
    const float* W2n     = (const float*)d_in[7];
    const float* b2      = (const float*)d_in[8];
    float*       out     = (float*)d_out;

    // Workspace: deg(50048) | agg(50000*256) | h(50000*256)  ~= 102.6 MB
    float* deg = (float*)d_ws;
    float* agg = deg + 50048;
    float* h   = agg + (size_t)N_NODES * IN_DIM;

    const int nAgg = N_NODES * IN_DIM;

    // --- degrees (computed once, reused for both layers) -------------------
    {
        int nz = 50048 + nAgg;   // zero deg + agg together
        zero_f32<<<(nz + 255) / 256, 256, 0, stream>>>(deg, nz);
    }
    degree_kernel<<<(N_EDGES + 255) / 256, 256, 0, stream>>>(edst, deg, N_EDGES);
    invdeg_kernel<<<(N_NODES + 255) / 256, 256, 0, stream>>>(deg, N_NODES);

    const int scatterThreads = N_EDGES * 64;         // 51.2M
    const int scatterBlocks  = (scatterThreads + 255) / 256;
    const dim3 blk(256);

    // --- layer 1: h = relu(in @ W1s + mean_agg(in) @ W1n + b1) -------------
    scatter_kernel<<<scatterBlocks, blk, 0, stream>>>(in_feat, esrc, edst, agg, N_EDGES);
    {
        dim3 grid(HID_DIM / 16, (3125 + 7) / 8);     // 16 x 391
        sage_gemm<HID_DIM, 1><<<grid, blk, 0, stream>>>(
            in_feat, agg, deg, W1s, W1n, b1, h);
    }

    // --- layer 2: out = h @ W2s + mean_agg(h) @ W2n + b2 -------------------
    zero_f32<<<(nAgg + 255) / 256, 256, 0, stream>>>(agg, nAgg);
    scatter_kernel<<<scatterBlocks, blk, 0, stream>>>(h, esrc, edst, agg, N_EDGES);
    {
        dim3 grid(OUT_DIM / 16, (3125 + 7) / 8);     // 8 x 391
        sage_gemm<OUT_DIM, 0><<<grid, blk, 0, stream>>>(
            h, agg, deg, W2s, W2n, b2, out);
    }
}